// DeformableTransformerACTR_66726611910726
// MI455X (gfx1250) — compile-verified
//
#include <hip/hip_runtime.h>
#include <hip/hip_bf16.h>

// ---------------------------------------------------------------------------
// Deformable Transformer decoder (fp32) for MI455X / gfx1250.
// GEMMs: V_WMMA_F32_16X16X4_F32 (full-fp32 tensor path), LDS-tiled with
// transposed B so every WMMA operand pair is one aligned ds_load_b64.
// Wave tile 32x64 (8 WMMAs per 6 LDS pair-loads), block tile 256x64.
// global_prefetch_b8 of the next k-chunk hides L2 latency of the stage.
// Sampling: one wave32 per (query, head), lane = channel -> each bilinear
// corner gather is a coalesced 128B read resident in the 192MB L2.
// ---------------------------------------------------------------------------

typedef float v2f __attribute__((ext_vector_type(2)));
typedef float v8f __attribute__((ext_vector_type(8)));

static constexpr int kB   = 2;
static constexpr int kNQ  = 8192;
static constexpr int kDM  = 256;
static constexpr int kDFF = 1024;
static constexpr int kNH  = 8;
static constexpr int kNL  = 4;
static constexpr int kNP  = 4;
static constexpr int kHD  = 32;
static constexpr int kLV  = 19560;          // 92*160 + 46*80 + 23*40 + 12*20
static constexpr int kLayers = 6;
static constexpr int kRows = kB * kNQ;      // 16384

// LDS pads: even so (row*pad + even k) stays 8B aligned for b64 pair loads;
// 34 = 2*17, 17 odd -> 34*n mod 64 distinct across 32 consecutive n (no bank conflicts).
static constexpr int kPad = 34;

// ---------------------------------------------------------------------------
// Prologue: flatten multi-scale feature maps [B,DM,H,W] -> src_flat [B,LV,DM]
// (and the padding mask -> mask_flat). Layer-invariant, runs once.
// ---------------------------------------------------------------------------
__global__ void flatten_kernel(const float* __restrict__ s0, const float* __restrict__ s1,
                               const float* __restrict__ s2, const float* __restrict__ s3,
                               const unsigned char* __restrict__ m0, const unsigned char* __restrict__ m1,
                               const unsigned char* __restrict__ m2, const unsigned char* __restrict__ m3,
                               float* __restrict__ srcf, unsigned char* __restrict__ maskf) {
    long i = (long)blockIdx.x * blockDim.x + threadIdx.x;     // over B*LV*DM
    long total = (long)kB * kLV * kDM;
    if (i >= total) return;
    int c = (int)(i % kDM);
    long bp = i / kDM;
    int pos = (int)(bp % kLV);
    int b = (int)(bp / kLV);
    const float* s; const unsigned char* mm; int hw, p;
    if (pos < 14720)      { s = s0; mm = m0; hw = 14720; p = pos; }
    else if (pos < 18400) { s = s1; mm = m1; hw = 3680;  p = pos - 14720; }
    else if (pos < 19320) { s = s2; mm = m2; hw = 920;   p = pos - 18400; }
    else                  { s = s3; mm = m3; hw = 240;   p = pos - 19320; }
    srcf[i] = s[((size_t)b * kDM + c) * hw + p];
    if (c == 0) maskf[(size_t)b * kLV + pos] = mm[(size_t)b * hw + p];
}

__global__ void valid_ratio_kernel(const unsigned char* __restrict__ m0, const unsigned char* __restrict__ m1,
                                   const unsigned char* __restrict__ m2, const unsigned char* __restrict__ m3,
                                   float* __restrict__ vr) {
    int t = threadIdx.x;
    if (t >= kB * kNL) return;
    int b = t >> 2, l = t & 3;
    const int Hs[4] = {92, 46, 23, 12};
    const int Ws[4] = {160, 80, 40, 20};
    const unsigned char* m = (l == 0) ? m0 : (l == 1) ? m1 : (l == 2) ? m2 : m3;
    int H = Hs[l], W = Ws[l];
    int ch = 0, cw = 0;
    for (int r = 0; r < H; ++r) ch += (m[((size_t)b * H + r) * W] == 0);
    for (int c = 0; c < W; ++c) cw += (m[(size_t)b * H * W + c] == 0);
    vr[(b * kNL + l) * 2 + 0] = (float)cw / (float)W;   // vW
    vr[(b * kNL + l) * 2 + 1] = (float)ch / (float)H;   // vH
}

__global__ void add_kernel(const float* __restrict__ a, const float* __restrict__ b,
                           float* __restrict__ o, long n) {
    long i = (long)blockIdx.x * blockDim.x + threadIdx.x;
    if (i < n) o[i] = a[i] + b[i];
}

// ---------------------------------------------------------------------------
// fp32 WMMA GEMM: C[M,N] = A[M,K] @ Bw[K,N] + bias[N]  (+relu / +row-mask)
// Block = 256 threads = 8 waves. Wave tile 32x64: c[2][4] accumulators; per
// k-step (K+=4): 2 A pair-loads + 4 B pair-loads feed 8 WMMAs. Block tile
// 256x64. A staged row-major [256][32], B staged TRANSPOSED [64][32] so the
// (k,k+1) operand pair of a column is one aligned ds_load_b64 (pad 34).
// WMMA operand layout per ISA 7.12.2:
//   A 16x4: lane<16 -> {K0,K1}, lane>=16 -> {K2,K3}; row = lane&15
//   B 4x16: VGPR0 = row K(2*half), VGPR1 = row K(2*half+1); col = lane&15
//   C 16x16: VGPR r -> row r + 8*half, col = lane&15
// ---------------------------------------------------------------------------
__global__ __launch_bounds__(256)
void gemm_f32_wmma(const float* __restrict__ A, const float* __restrict__ Bw,
                   const float* __restrict__ bias, const unsigned char* __restrict__ rowmask,
                   float* __restrict__ C, int M, int N, int K, int relu) {
    __shared__ float As[256 * kPad];    // 256 rows x 32 k   (pad 34) ~34.8KB
    __shared__ float BsT[64 * kPad];    // 64 cols x 32 k    (pad 34) ~8.7KB

    const int tid  = threadIdx.x;
    const int wave = tid >> 5;
    const int lane = tid & 31;
    const int half = lane >> 4;
    const int m    = lane & 15;

    const int rowBase0 = blockIdx.y * 256;
    const int rowBase  = rowBase0 + wave * 32;
    const int colBase  = blockIdx.x * 64;

    v8f c[2][4] = {};

    for (int k0 = 0; k0 < K; k0 += 32) {
        // stage A tile (coalesced along K: 128B rows)
        for (int i = tid; i < 256 * 32; i += 256) {
            int r = i >> 5, kk = i & 31;
            int gr = rowBase0 + r; if (gr >= M) gr = M - 1;
            As[r * kPad + kk] = A[(size_t)gr * K + k0 + kk];
        }
        // stage B tile transposed (global read coalesced along N)
        for (int i = tid; i < 32 * 64; i += 256) {
            int kk = i >> 6, n = i & 63;
            BsT[n * kPad + kk] = Bw[(size_t)(k0 + kk) * N + colBase + n];
        }
        // prefetch next k-chunk while this one is consumed
        if (k0 + 32 < K) {
            int r = tid;                                   // 256 rows -> one line each
            int gr = rowBase0 + r; if (gr >= M) gr = M - 1;
            __builtin_prefetch(&A[(size_t)gr * K + k0 + 32], 0, 0);
            if (tid < 32)
                __builtin_prefetch(&Bw[(size_t)(k0 + 32 + tid) * N + colBase], 0, 0);
        }
        __syncthreads();

        #pragma unroll
        for (int kk = 0; kk < 32; kk += 4) {
            const int ka = kk + 2 * half;                  // even -> 8B aligned pairs
            v2f av[2], bv[4];
            av[0] = *(const v2f*)&As[(wave * 32 + m) * kPad + ka];
            av[1] = *(const v2f*)&As[(wave * 32 + 16 + m) * kPad + ka];
            #pragma unroll
            for (int j = 0; j < 4; ++j)
                bv[j] = *(const v2f*)&BsT[(j * 16 + m) * kPad + ka];
            #pragma unroll
            for (int j = 0; j < 4; ++j) {
                c[0][j] = __builtin_amdgcn_wmma_f32_16x16x4_f32(
                    false, av[0], false, bv[j], (short)0, c[0][j], false, false);
                c[1][j] = __builtin_amdgcn_wmma_f32_16x16x4_f32(
                    false, av[1], false, bv[j], (short)0, c[1][j], false, false);
            }
        }
        __syncthreads();
    }

    #pragma unroll
    for (int j = 0; j < 4; ++j) {
        const int col = colBase + j * 16 + m;
        const float bb = bias[col];
        #pragma unroll
        for (int mi = 0; mi < 2; ++mi) {
            #pragma unroll
            for (int r = 0; r < 8; ++r) {
                const int grow = rowBase + mi * 16 + r + 8 * half;
                if (grow < M) {
                    float v = c[mi][j][r] + bb;
                    if (relu) v = fmaxf(v, 0.0f);
                    if (rowmask && rowmask[grow]) v = 0.0f;
                    C[(size_t)grow * N + col] = v;
                }
            }
        }
    }
}

// ---------------------------------------------------------------------------
// Softmax over NL*NP = 16 attention logits per (b, q, head). In place.
// ---------------------------------------------------------------------------
__global__ void softmax16_kernel(float* __restrict__ att) {
    int t = blockIdx.x * blockDim.x + threadIdx.x;    // over B*NQ*NH
    if (t >= kRows * kNH) return;
    int row = t >> 3, h = t & 7;
    float* p = att + (size_t)row * (kNH * kNL * kNP) + h * 16;
    float mx = p[0];
    #pragma unroll
    for (int i = 1; i < 16; ++i) mx = fmaxf(mx, p[i]);
    float e[16], s = 0.0f;
    #pragma unroll
    for (int i = 0; i < 16; ++i) { e[i] = __expf(p[i] - mx); s += e[i]; }
    float inv = 1.0f / s;
    #pragma unroll
    for (int i = 0; i < 16; ++i) p[i] = e[i] * inv;
}

// ---------------------------------------------------------------------------
// Deformable bilinear sampling + attention-weighted accumulation.
// One wave32 per (b, q, head); lane = channel d (HD == 32). The 4-corner
// gathers are 128B contiguous per wave -> perfect coalescing, L2-resident.
// ---------------------------------------------------------------------------
__device__ __forceinline__ float bilin_corner(const float* __restrict__ val, int b, int lstart,
                                              int Wl, int Hl, int xi, int yi, float w, int col) {
    if (w == 0.0f || xi < 0 || yi < 0 || xi >= Wl || yi >= Hl) return 0.0f;
    size_t idx = (size_t)b * kLV + lstart + (size_t)yi * Wl + xi;
    return w * val[idx * kDM + col];
}

__global__ __launch_bounds__(256)
void deform_sample_kernel(const float* __restrict__ val, const float* __restrict__ off,
                          const float* __restrict__ aw, const float* __restrict__ qref,
                          const float* __restrict__ vr, float* __restrict__ acc) {
    const int Hs[4] = {92, 46, 23, 12};
    const int Ws[4] = {160, 80, 40, 20};
    const int St[4] = {0, 14720, 18400, 19320};

    int wave = threadIdx.x >> 5;
    int d    = threadIdx.x & 31;
    int g    = blockIdx.x * 8 + wave;          // (b, q, h) flat
    int b    = g / (kNQ * kNH);
    int rem  = g - b * (kNQ * kNH);
    int qi   = rem / kNH;
    int h    = rem - qi * kNH;

    size_t row = (size_t)b * kNQ + qi;
    float rx0 = qref[row * 2 + 0];
    float ry0 = qref[row * 2 + 1];
    int col = h * kHD + d;
    float a = 0.0f;

    #pragma unroll
    for (int l = 0; l < 4; ++l) {
        const float Wl = (float)Ws[l], Hl = (float)Hs[l];
        float vrx = vr[(b * kNL + l) * 2 + 0];
        float vry = vr[(b * kNL + l) * 2 + 1];
        // x = (ref*vr + off/W)*W - 0.5 = ref*vr*W + off - 0.5
        float bx = rx0 * vrx * Wl - 0.5f;
        float by = ry0 * vry * Hl - 0.5f;
        #pragma unroll
        for (int p = 0; p < 4; ++p) {
            int oc = ((h * kNL + l) * kNP + p) * 2;
            float x = bx + off[row * 256 + oc];
            float y = by + off[row * 256 + oc + 1];
            float w_a = aw[row * 128 + h * 16 + l * 4 + p];
            float x0f = floorf(x), y0f = floorf(y);
            float lx = x - x0f, ly = y - y0f;
            int x0 = (int)x0f, y0 = (int)y0f;
            float w00 = (1.0f - lx) * (1.0f - ly);
            float w10 = lx * (1.0f - ly);
            float w01 = (1.0f - lx) * ly;
            float w11 = lx * ly;
            float s = bilin_corner(val, b, St[l], Ws[l], Hs[l], x0,     y0,     w00, col)
                    + bilin_corner(val, b, St[l], Ws[l], Hs[l], x0 + 1, y0,     w10, col)
                    + bilin_corner(val, b, St[l], Ws[l], Hs[l], x0,     y0 + 1, w01, col)
                    + bilin_corner(val, b, St[l], Ws[l], Hs[l], x0 + 1, y0 + 1, w11, col);
            a += w_a * s;
        }
    }
    acc[row * kDM + col] = a;
}

// ---------------------------------------------------------------------------
// Fused residual add + LayerNorm over DM=256. Block per row, thread per col.
// ---------------------------------------------------------------------------
__global__ __launch_bounds__(256)
void add_ln_kernel(const float* __restrict__ res, const float* __restrict__ y,
                   const float* __restrict__ g, const float* __restrict__ be,
                   float* __restrict__ out) {
    __shared__ float red[256];
    int row = blockIdx.x;
    int c = threadIdx.x;
    size_t base = (size_t)row * kDM;
    float v = res[base + c] + y[base + c];
    red[c] = v;
    __syncthreads();
    for (int s = 128; s > 0; s >>= 1) { if (c < s) red[c] += red[c + s]; __syncthreads(); }
    float mu = red[0] * (1.0f / kDM);
    __syncthreads();
    float dv = v - mu;
    red[c] = dv * dv;
    __syncthreads();
    for (int s = 128; s > 0; s >>= 1) { if (c < s) red[c] += red[c + s]; __syncthreads(); }
    float var = red[0] * (1.0f / kDM);
    out[base + c] = dv * rsqrtf(var + 1e-5f) * g[c] + be[c];
}

// ---------------------------------------------------------------------------
// Host-side launch sequence (graph-capture safe: kernels + async d2d only).
// ---------------------------------------------------------------------------
extern "C" void kernel_launch(void* const* d_in, const int* in_sizes, int n_in,
                              void* d_out, int out_size, void* d_ws, size_t ws_size,
                              hipStream_t stream) {
    (void)in_sizes; (void)n_in; (void)out_size; (void)ws_size;
    // input order: src0,mask0,src1,mask1,src2,mask2,src3,mask3,
    //              q_feat,q_pos,q_ref, W_off,b_off,W_attn,b_attn,W_val,b_val,
    //              W_out,b_out,W1,b1,W2,b2, g1,be1,g2,be2
    const float* src[4]        = {(const float*)d_in[0], (const float*)d_in[2],
                                  (const float*)d_in[4], (const float*)d_in[6]};
    const unsigned char* msk[4]= {(const unsigned char*)d_in[1], (const unsigned char*)d_in[3],
                                  (const unsigned char*)d_in[5], (const unsigned char*)d_in[7]};
    const float* q_feat = (const float*)d_in[8];
    const float* q_pos  = (const float*)d_in[9];
    const float* q_ref  = (const float*)d_in[10];
    const float* W_off  = (const float*)d_in[11];
    const float* b_off  = (const float*)d_in[12];
    const float* W_attn = (const float*)d_in[13];
    const float* b_attn = (const float*)d_in[14];
    const float* W_val  = (const float*)d_in[15];
    const float* b_val  = (const float*)d_in[16];
    const float* W_out  = (const float*)d_in[17];
    const float* b_out  = (const float*)d_in[18];
    const float* W1     = (const float*)d_in[19];
    const float* b1     = (const float*)d_in[20];
    const float* W2     = (const float*)d_in[21];
    const float* b2     = (const float*)d_in[22];
    const float* g1     = (const float*)d_in[23];
    const float* be1    = (const float*)d_in[24];
    const float* g2     = (const float*)d_in[25];
    const float* be2    = (const float*)d_in[26];
    float* out = (float*)d_out;

    // workspace carve (~224 MB)
    char* w = (char*)d_ws;
    auto carve = [&](size_t bytes) -> void* {
        void* p = (void*)w;
        w += (bytes + 255) & ~(size_t)255;
        return p;
    };
    float* SRCF = (float*)carve((size_t)kB * kLV * kDM * 4);   // 40 MB
    unsigned char* MASKF = (unsigned char*)carve((size_t)kB * kLV);
    float* VR   = (float*)carve((size_t)kB * kNL * 2 * 4);
    float* Q    = (float*)carve((size_t)kRows * kDM * 4);      // qf (ping)
    float* QP   = (float*)carve((size_t)kRows * kDM * 4);      // q = qf+pos, later acc
    float* OFF  = (float*)carve((size_t)kRows * kDM * 4);
    float* ATT  = (float*)carve((size_t)kRows * 128 * 4);
    float* VAL  = (float*)carve((size_t)kB * kLV * kDM * 4);   // 40 MB
    float* S2   = (float*)carve((size_t)kRows * kDM * 4);      // src2 / ffn out
    float* HB   = (float*)carve((size_t)kRows * kDFF * 4);     // 67 MB

    // prologue (layer invariant)
    {
        long total = (long)kB * kLV * kDM;
        int blocks = (int)((total + 255) / 256);
        flatten_kernel<<<blocks, 256, 0, stream>>>(src[0], src[1], src[2], src[3],
                                                   msk[0], msk[1], msk[2], msk[3], SRCF, MASKF);
        valid_ratio_kernel<<<1, 32, 0, stream>>>(msk[0], msk[1], msk[2], msk[3], VR);
        hipMemcpyAsync(Q, q_feat, (size_t)kRows * kDM * 4, hipMemcpyDeviceToDevice, stream);
    }

    const long nElem = (long)kRows * kDM;
    const int  addBlocks = (int)((nElem + 255) / 256);
    const dim3 gemmBlk(256);
    const int  gyQ = (kRows + 255) / 256;                  // 64
    const int  gyV = (kB * kLV + 255) / 256;               // 153

    for (int i = 0; i < kLayers; ++i) {
        const float* wOff  = W_off  + (size_t)i * kDM * 256;
        const float* bOff  = b_off  + (size_t)i * 256;
        const float* wAtt  = W_attn + (size_t)i * kDM * 128;
        const float* bAtt  = b_attn + (size_t)i * 128;
        const float* wVal  = W_val  + (size_t)i * kDM * kDM;
        const float* bVal  = b_val  + (size_t)i * kDM;
        const float* wOut  = W_out  + (size_t)i * kDM * kDM;
        const float* bOut  = b_out  + (size_t)i * kDM;
        const float* w1    = W1     + (size_t)i * kDM * kDFF;
        const float* bb1   = b1     + (size_t)i * kDFF;
        const float* w2    = W2     + (size_t)i * kDFF * kDM;
        const float* bb2   = b2     + (size_t)i * kDM;

        // q = qf + q_pos
        add_kernel<<<addBlocks, 256, 0, stream>>>(Q, q_pos, QP, nElem);

        // sampling offsets + attention logits
        gemm_f32_wmma<<<dim3(256 / 64, gyQ), gemmBlk, 0, stream>>>(
            QP, wOff, bOff, nullptr, OFF, kRows, 256, kDM, 0);
        gemm_f32_wmma<<<dim3(128 / 64, gyQ), gemmBlk, 0, stream>>>(
            QP, wAtt, bAtt, nullptr, ATT, kRows, 128, kDM, 0);
        softmax16_kernel<<<(kRows * kNH + 255) / 256, 256, 0, stream>>>(ATT);

        // value projection over all multi-scale positions (masked rows -> 0)
        gemm_f32_wmma<<<dim3(kDM / 64, gyV), gemmBlk, 0, stream>>>(
            SRCF, wVal, bVal, MASKF, VAL, kB * kLV, kDM, kDM, 0);

        // deformable attention gather (acc reuses QP)
        deform_sample_kernel<<<(kRows * kNH) / 8, 256, 0, stream>>>(
            VAL, OFF, ATT, q_ref, VR, QP);

        // output projection + residual LN
        gemm_f32_wmma<<<dim3(kDM / 64, gyQ), gemmBlk, 0, stream>>>(
            QP, wOut, bOut, nullptr, S2, kRows, kDM, kDM, 0);
        add_ln_kernel<<<kRows, 256, 0, stream>>>(Q, S2, g1 + (size_t)i * kDM,
                                                 be1 + (size_t)i * kDM, Q);

        // FFN
        gemm_f32_wmma<<<dim3(kDFF / 64, gyQ), gemmBlk, 0, stream>>>(
            Q, w1, bb1, nullptr, HB, kRows, kDFF, kDM, 1);
        gemm_f32_wmma<<<dim3(kDM / 64, gyQ), gemmBlk, 0, stream>>>(
            HB, w2, bb2, nullptr, S2, kRows, kDM, kDFF, 0);
        float* lnOut = (i == kLayers - 1) ? out : Q;
        add_ln_kernel<<<kRows, 256, 0, stream>>>(Q, S2, g2 + (size_t)i * kDM,
                                                 be2 + (size_t)i * kDM, lnOut);
    }
}